// Model_16174846837038
// MI455X (gfx1250) — compile-verified
//
#include <hip/hip_runtime.h>

// Continuous-time leaky RNN, forward Euler.  alpha = DT/TAU = 0.2
#define ALPHA 0.2f
#define BETA  0.8f
#define Bsz   64
#define Tsz   2048
#define INsz  64
#define Hsz   256

typedef float v2f __attribute__((ext_vector_type(2)));
typedef float v8f __attribute__((ext_vector_type(8)));

// Hardware TANH trans op (gfx1250) with compile-safe fallback.
#if __has_builtin(__builtin_amdgcn_tanhf)
  #define FAST_TANH(x) __builtin_amdgcn_tanhf(x)
#elif __has_builtin(__builtin_amdgcn_tanh_f32)
  #define FAST_TANH(x) __builtin_amdgcn_tanh_f32(x)
#else
  #define FAST_TANH(x) tanhf(x)
#endif

// -------------------------------------------------------------------------
// Kernel 1: xp[b,t,n] = ALPHA * (I[b,t,:] @ W_in[n,:] + bvec[n])
// Written into the u-region of d_out; the recurrence kernel consumes each
// element exactly once before overwriting it with u[b,t,n].
// One wave = one 16x16 (M over B*T, N over H) tile, K=64 via 16
// V_WMMA_F32_16X16X4_F32 in two independent accumulator chains.
// -------------------------------------------------------------------------
__global__ __launch_bounds__(256) void xproj_kernel(
    const float* __restrict__ I, const float* __restrict__ Win,
    const float* __restrict__ bvec, float* __restrict__ uout)
{
  const int lane = threadIdx.x & 31;
  const int wave = threadIdx.x >> 5;
  const int lo = lane & 15, hi = lane >> 4;

  const int tileId = blockIdx.x * 8 + wave;   // 8192 M-tiles * 16 N-tiles
  const int ntile  = tileId & 15;
  const int mtile  = tileId >> 4;
  const int r0 = mtile * 16;
  const int n  = ntile * 16 + lo;

  // A fragment: lane -> row M = lo, K pair (4c + 2*hi, +1)   [ISA 7.12.2]
  // B fragment: lane -> col N = lo, K pair (4c + 2*hi, +1)
  const float* Arow = I   + (size_t)(r0 + lo) * INsz;
  const float* Brow = Win + (size_t)n * INsz;          // W_in^T[k][n] = W_in[n][k]

  v8f acc0 = {}, acc1 = {};
  #pragma unroll
  for (int c8 = 0; c8 < INsz; c8 += 8) {               // 2 chains of 8 WMMAs
    const int k0 = c8 + 2 * hi;
    v2f a0 = *(const v2f*)(Arow + k0);
    v2f b0 = *(const v2f*)(Brow + k0);
    acc0 = __builtin_amdgcn_wmma_f32_16x16x4_f32(false, a0, false, b0,
                                                 (short)0, acc0, false, false);
    v2f a1 = *(const v2f*)(Arow + k0 + 4);
    v2f b1 = *(const v2f*)(Brow + k0 + 4);
    acc1 = __builtin_amdgcn_wmma_f32_16x16x4_f32(false, a1, false, b1,
                                                 (short)0, acc1, false, false);
  }

  const float bb = bvec[n];
  #pragma unroll
  for (int j = 0; j < 8; ++j) {                        // C: vgpr j -> M = j + 8*hi
    const int row = r0 + j + 8 * hi;
    uout[(size_t)row * Hsz + n] = ALPHA * (acc0[j] + acc1[j] + bb);
  }
}

// -------------------------------------------------------------------------
// Kernel 2: persistent recurrence. 4 workgroups x 512 threads (16 waves).
// WG g owns batch rows [16g, 16g+16); wave w owns H columns [16w, 16w+16).
// u tile lives in registers (WMMA C layout). W_rec^T fragments are
// pre-swizzled into LDS (256 KB) so every B operand is one ds_load_b64.
// r = tanh(u) is exchanged through a padded LDS tile each step.
// No inter-workgroup sync: batch rows are independent.
// 4 accumulator chains -> 4-way ILP in the XDL pipe on the serial path.
// -------------------------------------------------------------------------
#define RSTRIDE 258   // even (b64-aligned A pairs), low bank conflicts
#define WTF_FLOATS (16 * 64 * 32 * 2)                 // 65536 floats = 256 KB
#define RNN_LDS_BYTES ((WTF_FLOATS + 16 * RSTRIDE) * 4)

__global__ __launch_bounds__(512) void rnn_kernel(
    const float* __restrict__ x0, const float* __restrict__ Wrec,
    float* __restrict__ uout)
{
  extern __shared__ float smem[];
  float* wtf = smem;                 // swizzled W_rec^T B-fragments
  float* rls = smem + WTF_FLOATS;    // r tile: [16][RSTRIDE]

  const int lane = threadIdx.x & 31;
  const int wave = threadIdx.x >> 5;       // == ntile, 0..15
  const int lo = lane & 15, hi = lane >> 4;
  const int b0 = blockIdx.x * 16;
  const int n  = wave * 16 + lo;

  // ---- stage W_rec^T B-fragments into LDS (one-time, reused 2048x) ----
  {
    const float* src = Wrec + (size_t)n * Hsz;         // row n of W_rec
    float* dst = wtf + ((size_t)(wave * 64) * 32 + lane) * 2;
    #pragma unroll 4
    for (int c = 0; c < 64; ++c) {
      const int kk = c * 4 + 2 * hi;
      *(v2f*)dst = *(const v2f*)(src + kk);
      dst += 64;                                       // 32 lanes * 2 floats
    }
  }

  // ---- u tile in registers, C layout: vgpr j <-> (M = j + 8*hi, N = n) ----
  float u[8];
  unsigned obase[8];
  #pragma unroll
  for (int j = 0; j < 8; ++j) {
    const int m = j + 8 * hi;
    u[j] = x0[(size_t)(b0 + m) * Hsz + n];
    obase[j] = (unsigned)((b0 + m) * Tsz) * Hsz + n;   // < 2^31, fits
  }

  __syncthreads();

  const float* bfrag = wtf + (size_t)wave * 64 * 64 + lane * 2;
  const float* afrag = rls + lo * RSTRIDE + 2 * hi;

  for (int t = 0; t < Tsz; ++t) {
    // Prefetch next step's xp tile (global_prefetch_b8); one step (~WMMA
    // chain latency) of cover for the HBM miss. At t = Tsz-1 the addresses
    // land at the start of the y region -> still inside d_out.
    #pragma unroll
    for (int j = 0; j < 8; ++j)
      __builtin_prefetch(&uout[(size_t)obase[j] + (size_t)(t + 1) * Hsz], 0, 1);

    // r = tanh(u) -> LDS (all waves need all 256 columns as A operands)
    #pragma unroll
    for (int j = 0; j < 8; ++j)
      rls[(j + 8 * hi) * RSTRIDE + n] = FAST_TANH(u[j]);
    __syncthreads();

    // xp tile (same addresses we overwrite with u below)
    float xp[8];
    #pragma unroll
    for (int j = 0; j < 8; ++j)
      xp[j] = uout[(size_t)obase[j] + (size_t)t * Hsz];

    // acc = r @ W_rec^T : 16x16 tile, K=256 -> 64 WMMAs in 4 chains of 16
    v8f acc0 = {}, acc1 = {}, acc2 = {}, acc3 = {};
    #pragma unroll 4
    for (int c = 0; c < 64; c += 4) {
      v2f a0 = *(const v2f*)(afrag + (c + 0) * 4);
      v2f b0 = *(const v2f*)(bfrag + (c + 0) * 64);
      acc0 = __builtin_amdgcn_wmma_f32_16x16x4_f32(false, a0, false, b0,
                                                   (short)0, acc0, false, false);
      v2f a1 = *(const v2f*)(afrag + (c + 1) * 4);
      v2f b1 = *(const v2f*)(bfrag + (c + 1) * 64);
      acc1 = __builtin_amdgcn_wmma_f32_16x16x4_f32(false, a1, false, b1,
                                                   (short)0, acc1, false, false);
      v2f a2 = *(const v2f*)(afrag + (c + 2) * 4);
      v2f b2 = *(const v2f*)(bfrag + (c + 2) * 64);
      acc2 = __builtin_amdgcn_wmma_f32_16x16x4_f32(false, a2, false, b2,
                                                   (short)0, acc2, false, false);
      v2f a3 = *(const v2f*)(afrag + (c + 3) * 4);
      v2f b3 = *(const v2f*)(bfrag + (c + 3) * 64);
      acc3 = __builtin_amdgcn_wmma_f32_16x16x4_f32(false, a3, false, b3,
                                                   (short)0, acc3, false, false);
    }

    // u <- BETA*u + ALPHA*acc + xp ; stream u out
    #pragma unroll
    for (int j = 0; j < 8; ++j) {
      const float s = (acc0[j] + acc1[j]) + (acc2[j] + acc3[j]);
      u[j] = BETA * u[j] + ALPHA * s + xp[j];
      __builtin_nontemporal_store(u[j], &uout[(size_t)obase[j] + (size_t)t * Hsz]);
    }
    __syncthreads();   // protect rls before next step's writes
  }
}

// -------------------------------------------------------------------------
// Kernel 3: y[row, o] = u[row,:] @ Wout[o,:] + bout[o], o in {0,1}.
// Memory-bound (streams 134 MB once); one wave per (b,t) row.
// -------------------------------------------------------------------------
__global__ __launch_bounds__(256) void readout_kernel(
    const float* __restrict__ u, const float* __restrict__ Wout,
    const float* __restrict__ bout, float* __restrict__ y)
{
  const int lane = threadIdx.x & 31;
  const int row  = blockIdx.x * 8 + (threadIdx.x >> 5);  // over B*T
  const float* ur = u + (size_t)row * Hsz + lane * 8;

  float s0 = 0.f, s1 = 0.f;
  #pragma unroll
  for (int i = 0; i < 8; ++i) {
    const float v = __builtin_nontemporal_load(ur + i);
    s0 += v * Wout[lane * 8 + i];
    s1 += v * Wout[Hsz + lane * 8 + i];
  }
  #pragma unroll
  for (int off = 16; off > 0; off >>= 1) {
    s0 += __shfl_down(s0, off, 32);
    s1 += __shfl_down(s1, off, 32);
  }
  if (lane == 0) {
    y[(size_t)row * 2 + 0] = s0 + bout[0];
    y[(size_t)row * 2 + 1] = s1 + bout[1];
  }
}

// -------------------------------------------------------------------------
extern "C" void kernel_launch(void* const* d_in, const int* in_sizes, int n_in,
                              void* d_out, int out_size, void* d_ws, size_t ws_size,
                              hipStream_t stream) {
  (void)in_sizes; (void)n_in; (void)out_size; (void)d_ws; (void)ws_size;
  const float* x0   = (const float*)d_in[0];   // [B,H]
  const float* I    = (const float*)d_in[1];   // [B,T,IN]
  const float* Win  = (const float*)d_in[2];   // [H,IN]
  const float* Wrec = (const float*)d_in[3];   // [H,H]
  const float* bvec = (const float*)d_in[4];   // [H]
  const float* Wout = (const float*)d_in[5];   // [2,H]
  const float* bout = (const float*)d_in[6];   // [2]

  float* uout = (float*)d_out;                           // [B,T,H]
  float* y    = uout + (size_t)Bsz * Tsz * Hsz;          // [B,T,2]

  // 1) xp = alpha*(I @ Win^T + b) into the u region (consumed in-place)
  xproj_kernel<<<(Bsz * Tsz / 16) * 16 / 8, 256, 0, stream>>>(I, Win, bvec, uout);

  // 2) persistent recurrence: 4 WGs (16 batch rows each), 16 waves each
  rnn_kernel<<<Bsz / 16, 512, RNN_LDS_BYTES, stream>>>(x0, Wrec, uout);

  // 3) readout
  readout_kernel<<<Bsz * Tsz / 8, 256, 0, stream>>>(uout, Wout, bout, y);
}